// AdaptiveMask_35124242546785
// MI455X (gfx1250) — compile-verified
//
#include <hip/hip_runtime.h>
#include <stdint.h>

// AdaptiveMask: B x 512 binary mask from 20 piecewise-linear "tents" per row.
// Bandwidth-bound (~19 MB traffic). gfx1250 paths used: async global->LDS
// staging (ASYNCcnt) + s_wait_asynccnt split-counter wait. WMMA is
// intentionally NOT used: the P-reduction has no shared matmul factorization.

static constexpr int PP      = 20;   // prototypes per row
static constexpr int MAXLEN  = 512;  // output positions per row
static constexpr int NROW    = 8;    // rows per workgroup (one wave32 each)
static constexpr int THREADS = 256;  // 8 waves

__device__ __forceinline__ uint32_t lds_byte_off(const void* p) {
  // Flat pointers to LDS carry the wave-relative LDS byte offset in addr[31:0]
  // (CDNA5 ISA 10.2 aperture rules); this is what async VDST expects.
  return (uint32_t)(uintptr_t)p;
}

__device__ __forceinline__ void async_load_f32_to_lds(uint32_t lds_off, const float* g) {
  // GV mode: LDS address is LDS base plus the VDST VGPR value; global addr from VGPR pair.
  asm volatile("global_load_async_to_lds_b32 %0, %1, off"
               :
               : "v"(lds_off), "v"(g)
               : "memory");
}

__device__ __forceinline__ void wait_async_zero() {
  asm volatile("s_wait_asynccnt 0" ::: "memory");
}

__global__ __launch_bounds__(THREADS)
void adaptive_mask_kernel(const float* __restrict__ tok,
                          const float* __restrict__ sigma,
                          const float* __restrict__ pi,
                          float* __restrict__ out,
                          int B) {
  // raw[row][field][p]: field 0=tok, 1=sigma, 2=pi
  __shared__ float  raw[NROW * 3 * PP];
  __shared__ float4 coefA[NROW][PP];   // {m, oneCL, sr, cRm1}
  __shared__ float  keepA[NROW][PP];

  const int tid = threadIdx.x;
  const int r0  = blockIdx.x * NROW;

  // ---- Phase 1: async-stage 8 rows x 60 floats into LDS ----
  for (int idx = tid; idx < NROW * 3 * PP; idx += THREADS) {
    int row   = idx / (3 * PP);
    int rem   = idx - row * (3 * PP);
    int field = rem / PP;
    int p     = rem - field * PP;
    if (r0 + row < B) {
      const float* src = (field == 0) ? tok : ((field == 1) ? sigma : pi);
      const float* g   = src + (size_t)(r0 + row) * PP + p;
      async_load_f32_to_lds(lds_byte_off(&raw[idx]), g);
    }
  }
  wait_async_zero();
  __syncthreads();

  const int wave = tid >> 5;
  const int lane = tid & 31;
  const int row  = wave;                       // one wave32 per row
  const bool row_ok = (r0 + row) < B;
  const float* rrow = &raw[row * 3 * PP];

  // ---- Phase 2: per-prototype coefficients (lanes 0..19) ----
  if (row_ok && lane < PP) {
    float s = 0.0f;                            // mean(pi) over the row
    for (int j = 0; j < PP; ++j) s += rrow[2 * PP + j];
    float mean = s * (1.0f / (float)PP);

    float t  = rrow[lane];
    float sg = rrow[PP + lane];
    float pv = rrow[2 * PP + lane];

    float m     = rintf(fminf(fmaxf(t, 1.0f), 511.0f));  // RNE == jnp.round
    float denom = (m == 511.0f) ? 1.0f : (511.0f - m);
    float sr    = (m - 512.0f) / denom;                  // right slope
    float oneCL = 1.0f + 0.001f * sg * m;                // left  = q + oneCL
    float cRm1  = 0.001f * sg * (512.0f - m) - 1.0f;     // right = q*sr + cRm1

    coefA[row][lane] = make_float4(m, oneCL, sr, cRm1);
    keepA[row][lane] = (pv >= mean) ? 1.0f : 0.0f;
  }
  __syncthreads();

  // ---- Phase 3: 512 outputs per row; 4 coalesced float4 stores per lane ----
  if (row_ok) {
    float* orow = out + (size_t)(r0 + row) * MAXLEN;
    const float inv3 = 1.0f / 3.0f;

    #pragma unroll
    for (int pass = 0; pass < 4; ++pass) {
      const int l0 = pass * 128 + lane * 4;
      float acc[4] = {0.0f, 0.0f, 0.0f, 0.0f};

      for (int p = 0; p < PP; ++p) {
        if (keepA[row][p] == 0.0f) continue;   // wave-uniform scalar branch
        const float4 c = coefA[row][p];        // ds_load_b128, broadcast
        #pragma unroll
        for (int j = 0; j < 4; ++j) {
          float lf    = (float)(l0 + j);
          float q     = lf - c.x;              // l - m
          float left  = q + c.y;
          float right = fmaf(q, c.z, c.w);
          float base  = (lf < c.x) ? left : right;
          acc[j] += fmaf(base, inv3, 1.0f);    // val = base/3 + 1
        }
      }

      float4 o;
      o.x = (acc[0] > 0.0f) ? 1.0f : 0.0f;
      o.y = (acc[1] > 0.0f) ? 1.0f : 0.0f;
      o.z = (acc[2] > 0.0f) ? 1.0f : 0.0f;
      o.w = (acc[3] > 0.0f) ? 1.0f : 0.0f;
      *(float4*)(orow + l0) = o;               // global_store_b128, coalesced
    }
  }
}

extern "C" void kernel_launch(void* const* d_in, const int* in_sizes, int n_in,
                              void* d_out, int out_size, void* d_ws, size_t ws_size,
                              hipStream_t stream) {
  const float* tok   = (const float*)d_in[0];  // all_selected_token_index [B,20]
  const float* sigma = (const float*)d_in[1];  // sigma [B,20]
  const float* pi    = (const float*)d_in[2];  // pi [B,20]
  float* out = (float*)d_out;                  // [B,512] f32

  const int B = in_sizes[0] / PP;
  const int grid = (B + NROW - 1) / NROW;
  adaptive_mask_kernel<<<grid, THREADS, 0, stream>>>(tok, sigma, pi, out, B);
}